// HierarchicalSparseLayer_56959856280362
// MI455X (gfx1250) — compile-verified
//
#include <hip/hip_runtime.h>
#include <hip/hip_bf16.h>
#include <math.h>

// ---------------------------------------------------------------------------
// HierarchicalSparseLayer for MI455X (gfx1250, wave32, WMMA)
// Routed top-2 MoE FFN + block-sparse cross layer, fp32 WMMA 16x16x4.
// 128x64 tiles, swizzled-B LDS, double-buffered async A staging,
// split accumulator chains, next-surviving-block prefetch.
// ---------------------------------------------------------------------------

typedef float v2f __attribute__((ext_vector_type(2)));
typedef float v8f __attribute__((ext_vector_type(8)));

static constexpr int B_TOK = 8192;
static constexpr int DIM   = 1024;   // D
static constexpr int HID   = 4096;   // H
static constexpr int NEXP  = 16;     // E
static constexpr int MTILE = 128;    // M rows per block tile
static constexpr int ROWCAP = 18432; // B*K (=16384) + 16*127 rounded up to 128

static constexpr int LDA  = 68;   // padded LDS stride for A tiles (floats)
static constexpr int LDBP = 160;  // LDS stride per k-PAIR row of swizzled B

// ---- workspace layout (bytes) ----
static constexpr size_t WS_COUNTS  = 0;                                   // 16 int
static constexpr size_t WS_CURSOR  = 64;                                  // 16 int
static constexpr size_t WS_OFFSETS = 128;                                 // 17 int
static constexpr size_t WS_TOKE    = 256;                                 // B*2 int
static constexpr size_t WS_TOKW    = WS_TOKE   + (size_t)B_TOK * 2 * 4;   // B*2 f32
static constexpr size_t WS_TOKPOS  = WS_TOKW   + (size_t)B_TOK * 2 * 4;   // B*2 int
static constexpr size_t WS_ROWTOK  = WS_TOKPOS + (size_t)B_TOK * 2 * 4;   // ROWCAP int
static constexpr size_t WS_ROWWT   = WS_ROWTOK + (size_t)ROWCAP * 4;      // ROWCAP f32
static constexpr size_t WS_YBUF    = (WS_ROWWT + (size_t)ROWCAP * 4 + 255) & ~(size_t)255;
static constexpr size_t WS_HBUF    = WS_YBUF + (size_t)ROWCAP * DIM * 4;  // ROWCAP*H f32

// ---------------------------------------------------------------------------
// CDNA5 async global->LDS copy (16B per lane, tracked by ASYNCcnt)
// ---------------------------------------------------------------------------
__device__ __forceinline__ void async_copy16(unsigned lds_off, const void* g) {
  asm volatile("global_load_async_to_lds_b128 %0, %1, off"
               :: "v"(lds_off), "v"((unsigned long long)(uintptr_t)g)
               : "memory");
}
template <int N>
__device__ __forceinline__ void wait_async() {
  asm volatile("s_wait_asynccnt %0" :: "i"(N) : "memory");
}
__device__ __forceinline__ unsigned lds_off32(const void* p) {
  return (unsigned)(uintptr_t)p;  // low 32 bits of LDS-aperture flat address
}

// ---------------------------------------------------------------------------
// WMMA helpers (fp32, 16x16x4).  Per CDNA5 ISA §7.12.2:
//   A 16x4:  lanes 0-15 row m=lane {K=k0,k0+1}; lanes 16-31 {K=k0+2,k0+3}
//   B 4x16:  lanes 0-15 col n=lane {K=k0,k0+1}; lanes 16-31 {K=k0+2,k0+3}
//   C/D:     vgpr j, lanes 0-15 -> (M=j, N=lane); lanes 16-31 -> (M=j+8)
// B is stored pair-interleaved in LDS: Bs[(k>>1)*LDBP + 2*n + (k&1)], so each
// fragment is one aligned ds_load_b64 and wave halves hit disjoint banks.
// ---------------------------------------------------------------------------
__device__ __forceinline__ v8f wmma_f32_4(v2f a, v2f b, v8f c) {
  return __builtin_amdgcn_wmma_f32_16x16x4_f32(false, a, false, b, (short)0, c,
                                               false, false);
}

__device__ __forceinline__ v2f frag_a(const float* Als, int mrow, int kk, int lane) {
  const int half = lane >> 4, l = lane & 15;
  return *(const v2f*)(Als + (mrow + l) * LDA + kk + half * 2);
}

__device__ __forceinline__ v2f frag_b(const float* Bs, int ncol, int kk, int lane) {
  const int half = lane >> 4, l = lane & 15;
  const int p = (kk >> 1) + half;           // k-pair row
  return *(const v2f*)(Bs + p * LDBP + 2 * (ncol + l));
}

// Stage one 64(k) x 64(n) dense weight block into swizzled LDS.
// Each thread: 2 float2 loads from consecutive k rows -> one float4 LDS store.
__device__ __forceinline__ void stage_b(float* Bs, const float* wB, size_t ldw,
                                        int kb, int nb, int tid) {
  const int col2 = (tid & 31) * 2;
  for (int q = tid >> 5; q < 32; q += 8) {
    const float2 g0 = *(const float2*)(wB + (size_t)(kb + 2 * q + 0) * ldw + nb + col2);
    const float2 g1 = *(const float2*)(wB + (size_t)(kb + 2 * q + 1) * ldw + nb + col2);
    float4 v; v.x = g0.x; v.y = g1.x; v.z = g0.y; v.w = g1.y;
    *(float4*)&Bs[q * LDBP + 2 * col2] = v;
  }
}

// ---------------------------------------------------------------------------
// k_init / k_gate / k_offsets / k_scatter: routing pipeline
// ---------------------------------------------------------------------------
__global__ void k_init(int* __restrict__ rowtok, int* __restrict__ counts,
                       int* __restrict__ cursor) {
  const int i = blockIdx.x * blockDim.x + threadIdx.x;
  if (i < ROWCAP) rowtok[i] = -1;
  if (i < NEXP) { counts[i] = 0; cursor[i] = 0; }
}

__global__ __launch_bounds__(256) void k_gate(
    const float* __restrict__ x, const float* __restrict__ gate_w,
    const float* __restrict__ temp, int* __restrict__ tok_e,
    float* __restrict__ tok_w, int* __restrict__ counts) {
  __shared__ float lg[16][17];
  const int tid = threadIdx.x;
  const int tl = tid >> 4;     // token slot in block
  const int e  = tid & 15;     // expert
  const int t  = blockIdx.x * 16 + tl;

  const float* gw = gate_w + (size_t)(2 - 1) * NEXP * DIM + (size_t)e * DIM;
  const float* xr = x + (size_t)t * DIM;
  float acc = 0.f;
  for (int d = 0; d < DIM; d += 4) {
    acc += xr[d + 0] * gw[d + 0] + xr[d + 1] * gw[d + 1] +
           xr[d + 2] * gw[d + 2] + xr[d + 3] * gw[d + 3];
  }
  lg[tl][e] = acc / temp[1];
  __syncthreads();

  if (e == 0) {
    float v0 = -INFINITY, v1 = -INFINITY;
    int i0 = 0, i1 = 0;
    for (int j = 0; j < NEXP; ++j) {
      const float v = lg[tl][j];
      if (v > v0)      { v1 = v0; i1 = i0; v0 = v; i0 = j; }
      else if (v > v1) { v1 = v; i1 = j; }
    }
    const float e1 = __expf(v1 - v0);
    const float inv = 1.f / (1.f + e1);
    tok_e[t * 2 + 0] = i0;  tok_e[t * 2 + 1] = i1;
    tok_w[t * 2 + 0] = inv; tok_w[t * 2 + 1] = e1 * inv;
    atomicAdd(&counts[i0], 1);
    atomicAdd(&counts[i1], 1);
  }
}

__global__ void k_offsets(const int* __restrict__ counts,
                          int* __restrict__ offsets, int* __restrict__ cursor) {
  if (threadIdx.x == 0) {
    int off = 0;
    for (int e = 0; e < NEXP; ++e) {
      offsets[e] = off;
      cursor[e]  = off;
      off += (counts[e] + MTILE - 1) & ~(MTILE - 1);
    }
    offsets[NEXP] = off;
  }
}

__global__ void k_scatter(const int* __restrict__ tok_e,
                          const float* __restrict__ tok_w,
                          int* __restrict__ cursor, int* __restrict__ rowtok,
                          float* __restrict__ rowwt, int* __restrict__ tok_pos) {
  const int t = blockIdx.x * blockDim.x + threadIdx.x;
  if (t >= B_TOK) return;
  for (int k = 0; k < 2; ++k) {
    const int e   = tok_e[t * 2 + k];
    const float w = tok_w[t * 2 + k];
    const int pos = atomicAdd(&cursor[e], 1);
    rowtok[pos] = t;
    rowwt[pos]  = w;
    tok_pos[t * 2 + k] = pos;
  }
}

// ---------------------------------------------------------------------------
// Shared GEMM core: 128x64 tile, 8 waves of 32x32, fp32 WMMA 16x16x4,
// two accumulation chains per output subtile (hide WMMA C->D latency).
// ---------------------------------------------------------------------------
#define GEMM_WAVE_SETUP()                                                     \
  const int lane = tid & 31, wave = tid >> 5;                                 \
  const int wm = wave & 3, wn = wave >> 2;                                    \
  const int mrow0 = wm * 32, mrow1 = wm * 32 + 16;                            \
  const int ncol0 = wn * 32, ncol1 = wn * 32 + 16;                            \
  v8f p00 = {}, p01 = {}, p10 = {}, p11 = {};                                 \
  v8f q00 = {}, q01 = {}, q10 = {}, q11 = {};

#define GEMM_KSTEPS(Ap, Bp)                                                   \
  for (int kk = 0; kk < 64; kk += 8) {                                        \
    {                                                                         \
      const v2f a0 = frag_a(Ap, mrow0, kk, lane);                             \
      const v2f a1 = frag_a(Ap, mrow1, kk, lane);                             \
      const v2f b0 = frag_b(Bp, ncol0, kk, lane);                             \
      const v2f b1 = frag_b(Bp, ncol1, kk, lane);                             \
      p00 = wmma_f32_4(a0, b0, p00);  p01 = wmma_f32_4(a0, b1, p01);          \
      p10 = wmma_f32_4(a1, b0, p10);  p11 = wmma_f32_4(a1, b1, p11);          \
    }                                                                         \
    {                                                                         \
      const v2f a0 = frag_a(Ap, mrow0, kk + 4, lane);                         \
      const v2f a1 = frag_a(Ap, mrow1, kk + 4, lane);                         \
      const v2f b0 = frag_b(Bp, ncol0, kk + 4, lane);                         \
      const v2f b1 = frag_b(Bp, ncol1, kk + 4, lane);                         \
      q00 = wmma_f32_4(a0, b0, q00);  q01 = wmma_f32_4(a0, b1, q01);          \
      q10 = wmma_f32_4(a1, b0, q10);  q11 = wmma_f32_4(a1, b1, q11);          \
    }                                                                         \
  }

#define GEMM_REDUCE()                                                         \
  const v8f acc00 = p00 + q00, acc01 = p01 + q01;                             \
  const v8f acc10 = p10 + q10, acc11 = p11 + q11;

// ---------------------------------------------------------------------------
// k_gemm1: Hbuf[row, nb:nb+64] = silu( gather(x) @ (w1[e]*mask1[e]) + b1[e] )
// ---------------------------------------------------------------------------
__global__ __launch_bounds__(256) void k_gemm1(
    const float* __restrict__ x, const float* __restrict__ w1,
    const float* __restrict__ b1, const unsigned char* __restrict__ mask1,
    const int* __restrict__ offsets, const int* __restrict__ rowtok,
    float* __restrict__ hbuf) {
  __shared__ float Als[2][MTILE * LDA];
  __shared__ float Bs[32 * LDBP];
  __shared__ int   ltok[MTILE];
  __shared__ int   soff[NEXP + 1];
  __shared__ int   klist[DIM / 64];
  __shared__ int   knum;

  const int tid = threadIdx.x;
  const int nb  = blockIdx.x * 64;  // H column base
  const int row0 = blockIdx.y * MTILE;

  if (tid <= NEXP) soff[tid] = offsets[tid];
  if (tid < MTILE) {
    const int tk = rowtok[row0 + tid];
    ltok[tid] = tk >= 0 ? tk : 0;   // padding rows gather token 0 (unused output)
  }
  __syncthreads();
  if (row0 >= soff[NEXP]) return;

  int e = 0;
  for (int i = 1; i < NEXP; ++i) if (row0 >= soff[i]) e = i;

  const float* w1e = w1 + (size_t)e * DIM * HID;
  const unsigned char* m1e = mask1 + (size_t)e * DIM * HID;

  if (tid == 0) {  // compact surviving (block-dense) k-blocks
    int n = 0;
    for (int b = 0; b < DIM / 64; ++b)
      if (m1e[(size_t)(b * 64) * HID + nb] != 0) klist[n++] = b * 64;
    knum = n;
  }
  __syncthreads();

  GEMM_WAVE_SETUP();
  const int rA = tid >> 4, cA = (tid & 15) * 4;

  auto issueA = [&](int buf, int kb) {  // 8 async b128 copies per wave
    for (int rr = rA; rr < MTILE; rr += 16)
      async_copy16(lds_off32(&Als[buf][rr * LDA + cA]),
                   x + (size_t)ltok[rr] * DIM + kb + cA);
  };

  const int kn = knum;
  if (kn > 0) issueA(0, klist[0]);
  for (int i = 0; i < kn; ++i) {
    const int cur = i & 1;
    __syncthreads();  // prev compute done: Als[cur^1] and Bs reusable
    if (i + 1 < kn) {
      issueA(cur ^ 1, klist[i + 1]);
      const int knx = klist[i + 1];  // prefetch next surviving weight block
      __builtin_prefetch((const char*)(w1e + (size_t)(knx + (tid >> 2)) * HID + nb) +
                         (tid & 3) * 64, 0, 0);
    }
    stage_b(Bs, w1e, HID, klist[i], nb, tid);
    if (i + 1 < kn) wait_async<8>();  // in-order: current block's copies done
    else            wait_async<0>();
    __syncthreads();
    GEMM_KSTEPS(&Als[cur][0], Bs);
  }
  GEMM_REDUCE();

  const int half = lane >> 4, l = lane & 15;
  const float* b1e = b1 + (size_t)e * HID;
  auto epi = [&](const v8f& acc, int mrow, int ncol) {
    for (int j = 0; j < 8; ++j) {
      const int m = mrow + j + half * 8;
      const int n = nb + ncol + l;
      const float v = acc[j] + b1e[n];
      hbuf[(size_t)(row0 + m) * HID + n] = v / (1.f + __expf(-v));
    }
  };
  epi(acc00, mrow0, ncol0); epi(acc01, mrow0, ncol1);
  epi(acc10, mrow1, ncol0); epi(acc11, mrow1, ncol1);
}

// ---------------------------------------------------------------------------
// k_gemm2: ybuf[row, nb:nb+64] = rowwt[row]*( hbuf[row] @ (w2[e]*mask2[e]) + b2[e] )
// ---------------------------------------------------------------------------
__global__ __launch_bounds__(256) void k_gemm2(
    const float* __restrict__ hbuf, const float* __restrict__ w2,
    const float* __restrict__ b2, const unsigned char* __restrict__ mask2,
    const int* __restrict__ offsets, const float* __restrict__ rowwt,
    float* __restrict__ ybuf) {
  __shared__ float Als[2][MTILE * LDA];
  __shared__ float Bs[32 * LDBP];
  __shared__ float lwt[MTILE];
  __shared__ int   soff[NEXP + 1];
  __shared__ int   klist[HID / 64];
  __shared__ int   knum;

  const int tid = threadIdx.x;
  const int nb  = blockIdx.x * 64;  // D column base
  const int row0 = blockIdx.y * MTILE;

  if (tid <= NEXP) soff[tid] = offsets[tid];
  if (tid < MTILE) lwt[tid] = rowwt[row0 + tid];
  __syncthreads();
  if (row0 >= soff[NEXP]) return;

  int e = 0;
  for (int i = 1; i < NEXP; ++i) if (row0 >= soff[i]) e = i;

  const float* w2e = w2 + (size_t)e * HID * DIM;
  const unsigned char* m2e = mask2 + (size_t)e * HID * DIM;

  if (tid == 0) {
    int n = 0;
    for (int b = 0; b < HID / 64; ++b)
      if (m2e[(size_t)(b * 64) * DIM + nb] != 0) klist[n++] = b * 64;
    knum = n;
  }
  __syncthreads();

  GEMM_WAVE_SETUP();
  const int rA = tid >> 4, cA = (tid & 15) * 4;

  auto issueA = [&](int buf, int kb) {
    for (int rr = rA; rr < MTILE; rr += 16)
      async_copy16(lds_off32(&Als[buf][rr * LDA + cA]),
                   hbuf + (size_t)(row0 + rr) * HID + kb + cA);
  };

  const int kn = knum;
  if (kn > 0) issueA(0, klist[0]);
  for (int i = 0; i < kn; ++i) {
    const int cur = i & 1;
    __syncthreads();
    if (i + 1 < kn) {
      issueA(cur ^ 1, klist[i + 1]);
      const int knx = klist[i + 1];
      __builtin_prefetch((const char*)(w2e + (size_t)(knx + (tid >> 2)) * DIM + nb) +
                         (tid & 3) * 64, 0, 0);
    }
    stage_b(Bs, w2e, DIM, klist[i], nb, tid);
    if (i + 1 < kn) wait_async<8>();
    else            wait_async<0>();
    __syncthreads();
    GEMM_KSTEPS(&Als[cur][0], Bs);
  }
  GEMM_REDUCE();

  const int half = lane >> 4, l = lane & 15;
  const float* b2e = b2 + (size_t)e * DIM;
  auto epi = [&](const v8f& acc, int mrow, int ncol) {
    for (int j = 0; j < 8; ++j) {
      const int m = mrow + j + half * 8;
      const int n = nb + ncol + l;
      ybuf[(size_t)(row0 + m) * DIM + n] = (acc[j] + b2e[n]) * lwt[m];
    }
  };
  epi(acc00, mrow0, ncol0); epi(acc01, mrow0, ncol1);
  epi(acc10, mrow1, ncol0); epi(acc11, mrow1, ncol1);
}

// ---------------------------------------------------------------------------
// k_gemm3: out = x + (ybuf[pos0]+ybuf[pos1]) @ (cross_w*cross_mask) + cross_b
// ---------------------------------------------------------------------------
__global__ __launch_bounds__(256) void k_gemm3(
    const float* __restrict__ x, const float* __restrict__ ybuf,
    const float* __restrict__ cross_w, const float* __restrict__ cross_b,
    const unsigned char* __restrict__ cmask, const int* __restrict__ tok_pos,
    float* __restrict__ out) {
  __shared__ float Als[MTILE * LDA];
  __shared__ float Bs[32 * LDBP];
  __shared__ int lp0[MTILE];
  __shared__ int lp1[MTILE];
  __shared__ int klist[DIM / 64];
  __shared__ int knum;

  const int tid = threadIdx.x;
  const int nb  = blockIdx.x * 64;  // D column base
  const int row0 = blockIdx.y * MTILE;

  if (tid < MTILE) {
    lp0[tid] = tok_pos[(row0 + tid) * 2 + 0];
    lp1[tid] = tok_pos[(row0 + tid) * 2 + 1];
  }
  if (tid == 0) {
    int n = 0;
    for (int b = 0; b < DIM / 64; ++b)
      if (cmask[(size_t)(b * 64) * DIM + nb] != 0) klist[n++] = b * 64;
    knum = n;
  }
  __syncthreads();

  GEMM_WAVE_SETUP();
  const int rA = tid >> 4, cA = (tid & 15) * 4;

  const int kn = knum;
  for (int i = 0; i < kn; ++i) {
    const int kb = klist[i];
    __syncthreads();
    for (int rr = rA; rr < MTILE; rr += 16) {  // stage A = expert combine
      const float4 a0 = *(const float4*)(ybuf + (size_t)lp0[rr] * DIM + kb + cA);
      const float4 a1 = *(const float4*)(ybuf + (size_t)lp1[rr] * DIM + kb + cA);
      float4 v;
      v.x = a0.x + a1.x; v.y = a0.y + a1.y;
      v.z = a0.z + a1.z; v.w = a0.w + a1.w;
      *(float4*)&Als[rr * LDA + cA] = v;
    }
    stage_b(Bs, cross_w, DIM, kb, nb, tid);
    if (i + 1 < kn) {
      const int knx = klist[i + 1];
      __builtin_prefetch((const char*)(cross_w + (size_t)(knx + (tid >> 2)) * DIM + nb) +
                         (tid & 3) * 64, 0, 0);
    }
    __syncthreads();
    GEMM_KSTEPS(Als, Bs);
  }
  GEMM_REDUCE();

  const int half = lane >> 4, l = lane & 15;
  auto epi = [&](const v8f& acc, int mrow, int ncol) {
    for (int j = 0; j < 8; ++j) {
      const int m = mrow + j + half * 8;
      const int n = nb + ncol + l;
      const size_t gi = (size_t)(row0 + m) * DIM + n;
      out[gi] = x[gi] + acc[j] + cross_b[n];
    }
  };
  epi(acc00, mrow0, ncol0); epi(acc01, mrow0, ncol1);
  epi(acc10, mrow1, ncol0); epi(acc11, mrow1, ncol1);
}

// ---------------------------------------------------------------------------
extern "C" void kernel_launch(void* const* d_in, const int* in_sizes, int n_in,
                              void* d_out, int out_size, void* d_ws,
                              size_t ws_size, hipStream_t stream) {
  (void)in_sizes; (void)n_in; (void)out_size; (void)ws_size;

  const float* x       = (const float*)d_in[0];
  const float* gate_w  = (const float*)d_in[1];
  const float* temp    = (const float*)d_in[2];
  const float* w1      = (const float*)d_in[3];
  const float* b1      = (const float*)d_in[4];
  const float* w2      = (const float*)d_in[5];
  const float* b2      = (const float*)d_in[6];
  const float* cross_w = (const float*)d_in[7];
  const float* cross_b = (const float*)d_in[8];
  const unsigned char* mask1 = (const unsigned char*)d_in[9];
  const unsigned char* mask2 = (const unsigned char*)d_in[10];
  const unsigned char* cmask = (const unsigned char*)d_in[11];
  float* out = (float*)d_out;

  char* ws = (char*)d_ws;
  int*   counts  = (int*)(ws + WS_COUNTS);
  int*   cursor  = (int*)(ws + WS_CURSOR);
  int*   offsets = (int*)(ws + WS_OFFSETS);
  int*   tok_e   = (int*)(ws + WS_TOKE);
  float* tok_w   = (float*)(ws + WS_TOKW);
  int*   tok_pos = (int*)(ws + WS_TOKPOS);
  int*   rowtok  = (int*)(ws + WS_ROWTOK);
  float* rowwt   = (float*)(ws + WS_ROWWT);
  float* ybuf    = (float*)(ws + WS_YBUF);
  float* hbuf    = (float*)(ws + WS_HBUF);

  k_init<<<(ROWCAP + 255) / 256, 256, 0, stream>>>(rowtok, counts, cursor);
  k_gate<<<B_TOK / 16, 256, 0, stream>>>(x, gate_w, temp, tok_e, tok_w, counts);
  k_offsets<<<1, 32, 0, stream>>>(counts, offsets, cursor);
  k_scatter<<<B_TOK / 256, 256, 0, stream>>>(tok_e, tok_w, cursor, rowtok,
                                             rowwt, tok_pos);
  k_gemm1<<<dim3(HID / 64, ROWCAP / MTILE), 256, 0, stream>>>(
      x, w1, b1, mask1, offsets, rowtok, hbuf);
  k_gemm2<<<dim3(DIM / 64, ROWCAP / MTILE), 256, 0, stream>>>(
      hbuf, w2, b2, mask2, offsets, rowwt, ybuf);
  k_gemm3<<<dim3(DIM / 64, B_TOK / MTILE), 256, 0, stream>>>(
      x, ybuf, cross_w, cross_b, cmask, tok_pos, out);
}